// PositionalInferenceBlock_5677946765981
// MI455X (gfx1250) — compile-verified
//
#include <hip/hip_runtime.h>
#include <hip/hip_bf16.h>

// ---------------------------------------------------------------------------
// T=64, S=32, F=256, A=64, P=32. All heavy math on v_wmma_f32_16x16x4_f32.
//   theta = x@Wa+ba (2048,64); x_phi = x@Wb[:256] (2048,64)  [bb dropped: softmax shift-inv]
//   x_g = x@Wg[:256]+bg (2048,256); posW = [pos0|pos1]@Wpos+bpos (1024,64)
//   posb = posW@WbP (1024,64); posg = posW@WgP (1024,256)
//   WtB=Wtmp@WbP (pad 32x64), WtG=Wtmp@WgP (pad 16x256), btG=btmp@WgP (256)
//   scores_s[i,j,k] = TXs[i][j,k] + Pterm[j][i,k]
//   scores_t[j,i,t] = Gterm[i][j,t] + temp[j,t,i,:]·thetaWtB[j,i,:]
//   out = attn_s@x_g (batch i) + attn_s@posg (batch j)
//       + attn_t@x_g (batch i_s) + attntmp@WtG + btG
// Every GEMM is tile-exact: M%16==0, N%32==0, K%8==0 -> branch-free inner loop.
// ---------------------------------------------------------------------------

typedef __attribute__((ext_vector_type(2))) float v2f;
typedef __attribute__((ext_vector_type(8))) float v8f;

#define WMMA_F32(a, b, c) __builtin_amdgcn_wmma_f32_16x16x4_f32( \
    false, (a), false, (b), (short)0, (c), false, false)

// Strided/batched f32 WMMA GEMM. One wave computes a 16x32 C tile.
//   C[bz][m,n] (+)= sum_k A[bz*bsA + m*rsA + k] * B[bz*bsB + k*rsB + n*csB]  (+bias[n])
// BKC=true : B k-contiguous (rsB==1, csB arbitrary)  -> b64 loads of B
// BKC=false: B n-contiguous (csB==1, rsB arbitrary)  -> strided scalar loads
// A is always k-contiguous (csA==1). Requires M%16==0, N%32==0, K%8==0.
template <bool BKC>
__global__ __launch_bounds__(32)
void gemm_wmma_f32(const float* __restrict__ A, const float* __restrict__ B,
                   float* __restrict__ C, const float* __restrict__ bias, int K,
                   int rsA, long bsA, int rsB, int csB, long bsB,
                   int ldC, long bsC, int accumulate)
{
    const long bz   = blockIdx.z;
    const int tile_m = blockIdx.y * 16;
    const int tile_n = blockIdx.x * 32;
    const int lane = threadIdx.x;
    const int hf   = lane >> 4;       // 0: lanes 0-15, 1: lanes 16-31
    const int l16  = lane & 15;
    const int m  = tile_m + l16;
    const int n0 = tile_n + l16;
    const int n1 = n0 + 16;

    // Per-lane base pointers (K striping: VGPR0 holds k=2*hf, VGPR1 holds k=2*hf+1)
    const float* pa = A + bz * bsA + (long)m * rsA + 2 * hf;
    const float* pb0;
    const float* pb1;
    if (BKC) {
        pb0 = B + bz * bsB + (long)n0 * csB + 2 * hf;
        pb1 = B + bz * bsB + (long)n1 * csB + 2 * hf;
    } else {
        pb0 = B + bz * bsB + (long)n0 + (long)(2 * hf) * rsB;
        pb1 = pb0 + 16;
    }

    v8f acc00 = {}, acc01 = {}, acc10 = {}, acc11 = {};

    for (int k0 = 0; k0 < K; k0 += 8) {
        v2f a0, a1, b00, b01, b10, b11;
        { float2 t = *(const float2*)(pa);     a0[0] = t.x; a0[1] = t.y; }
        { float2 t = *(const float2*)(pa + 4); a1[0] = t.x; a1[1] = t.y; }
        if (BKC) {
            { float2 t = *(const float2*)(pb0);     b00[0] = t.x; b00[1] = t.y; }
            { float2 t = *(const float2*)(pb0 + 4); b01[0] = t.x; b01[1] = t.y; }
            { float2 t = *(const float2*)(pb1);     b10[0] = t.x; b10[1] = t.y; }
            { float2 t = *(const float2*)(pb1 + 4); b11[0] = t.x; b11[1] = t.y; }
            pb0 += 8; pb1 += 8;
        } else {
            b00[0] = pb0[0];              b00[1] = pb0[rsB];
            b01[0] = pb0[(long)4 * rsB];  b01[1] = pb0[(long)5 * rsB];
            b10[0] = pb1[0];              b10[1] = pb1[rsB];
            b11[0] = pb1[(long)4 * rsB];  b11[1] = pb1[(long)5 * rsB];
            pb0 += (long)8 * rsB; pb1 += (long)8 * rsB;
        }
        acc00 = WMMA_F32(a0, b00, acc00);
        acc10 = WMMA_F32(a0, b10, acc10);
        acc01 = WMMA_F32(a1, b01, acc01);
        acc11 = WMMA_F32(a1, b11, acc11);
        pa += 8;
    }

    v8f d0 = acc00 + acc01;
    v8f d1 = acc10 + acc11;

    const float bv0 = bias ? bias[n0] : 0.0f;
    const float bv1 = bias ? bias[n1] : 0.0f;
    float* Cb = C + bz * bsC;
#pragma unroll
    for (int v = 0; v < 8; ++v) {
        const int cm = tile_m + v + 8 * hf;   // C layout: VGPR v -> row v (+8 for hi half)
        const long i0 = (long)cm * ldC + n0;
        const long i1 = (long)cm * ldC + n1;
        float o0 = d0[v] + bv0, o1 = d1[v] + bv1;
        if (accumulate) { o0 += Cb[i0]; o1 += Cb[i1]; }
        Cb[i0] = o0; Cb[i1] = o1;
    }
}

// posW[(j*32+k)][p*32+n] = sum_d positions[p,j,k,d]*Wpos[d,n] + bpos[n]  (K=9, tiny)
__global__ void pos_project(const float* __restrict__ posi, const float* __restrict__ Wpos,
                            const float* __restrict__ bpos, float* __restrict__ posW)
{
    const int idx = blockIdx.x * 256 + threadIdx.x;  // 65536 outputs
    const int n   = idx & 31;
    const int row = (idx >> 5) & 1023;
    const int p   = idx >> 15;
    float s = bpos[n];
    const float* src = posi + ((long)p * 1024 + row) * 9;
#pragma unroll
    for (int d = 0; d < 9; ++d) s += src[d] * Wpos[d * 32 + n];
    posW[(long)row * 64 + p * 32 + n] = s;
}

// WtB (32x64, rows>=10 zero) = Wtmp@WbP ; WtG (16x256, rows>=10 zero) = Wtmp@WgP ; btG = btmp@WgP
__global__ void precompute_w(const float* __restrict__ Wtmp, const float* __restrict__ btmp,
                             const float* __restrict__ Wb, const float* __restrict__ Wg,
                             float* __restrict__ WtB, float* __restrict__ WtG,
                             float* __restrict__ btG)
{
    const float* WbP = Wb + 256 * 64;    // (64,64)
    const float* WgP = Wg + 256 * 256;   // (64,256)
    const int t = threadIdx.x;           // 256 threads

    for (int idx = t; idx < 32 * 64; idx += 256) {
        const int d = idx >> 6, a = idx & 63;
        float s = 0.0f;
        if (d < 10)
            for (int mm = 0; mm < 64; ++mm) s += Wtmp[d * 64 + mm] * WbP[mm * 64 + a];
        WtB[idx] = s;
    }
    for (int idx = t; idx < 16 * 256; idx += 256) {
        const int d = idx >> 8, f = idx & 255;
        float s = 0.0f;
        if (d < 10)
            for (int mm = 0; mm < 64; ++mm) s += Wtmp[d * 64 + mm] * WgP[mm * 256 + f];
        WtG[idx] = s;
    }
    {
        float s = 0.0f;
        for (int mm = 0; mm < 64; ++mm) s += btmp[mm] * WgP[mm * 256 + t];
        btG[t] = s;
    }
}

// attn_s[i][j][k] = softmax_k( TXs[i][j][k] + Pterm[j][i][k] ); one wave per (i,j)
__global__ __launch_bounds__(32)
void softmax_s(const float* __restrict__ TXs, const float* __restrict__ Pterm,
               float* __restrict__ attn_s)
{
    const int ij = blockIdx.x;           // i*32+j : i in T=64, j in S=32
    const int i = ij >> 5, j = ij & 31;
    const int k = threadIdx.x;

    float s = TXs[(long)i * 1024 + j * 32 + k] + Pterm[(long)j * 2048 + i * 32 + k];
    float mx = s;
    for (int off = 16; off; off >>= 1) mx = fmaxf(mx, __shfl_xor(mx, off, 32));
    float e = __expf(s - mx);
    float sum = e;
    for (int off = 16; off; off >>= 1) sum += __shfl_xor(sum, off, 32);
    attn_s[(long)i * 1024 + j * 32 + k] = e / sum;
}

// scores_t[j,i,t] = Gterm[i][j][t] + sum_d temp[j,t,i,d]*thetaWtB[ji][d]
// -> attn_t[ji][t] and attntmp[ji][d] = sum_t attn*temp  (cols 10..15 zeroed)
__global__ __launch_bounds__(32)
void softmax_t(const float* __restrict__ Gterm, const float* __restrict__ temp,
               const float* __restrict__ thetaWtB,
               float* __restrict__ attn_t, float* __restrict__ attntmp)
{
    const int ji = blockIdx.x;           // j*32+i : j in T=64, i in S=32
    const int j = ji >> 5, i = ji & 31;
    const int lane = threadIdx.x;
    const int t0 = lane, t1 = lane + 32;

    float th[10];
#pragma unroll
    for (int d = 0; d < 10; ++d) th[d] = thetaWtB[(long)ji * 32 + d];

    const float* G = Gterm + ((long)i * 64 + j) * 64;
    float s0 = G[t0], s1 = G[t1];

    const float* tp0 = temp + ((long)(j * 64 + t0) * 32 + i) * 10;
    const float* tp1 = temp + ((long)(j * 64 + t1) * 32 + i) * 10;
    float tv0[10], tv1[10];
#pragma unroll
    for (int d = 0; d < 10; ++d) { tv0[d] = tp0[d]; tv1[d] = tp1[d]; }
#pragma unroll
    for (int d = 0; d < 10; ++d) { s0 += tv0[d] * th[d]; s1 += tv1[d] * th[d]; }

    float mx = fmaxf(s0, s1);
    for (int off = 16; off; off >>= 1) mx = fmaxf(mx, __shfl_xor(mx, off, 32));
    float e0 = __expf(s0 - mx), e1 = __expf(s1 - mx);
    float sum = e0 + e1;
    for (int off = 16; off; off >>= 1) sum += __shfl_xor(sum, off, 32);
    const float inv = 1.0f / sum;
    e0 *= inv; e1 *= inv;

    attn_t[(long)ji * 64 + t0] = e0;
    attn_t[(long)ji * 64 + t1] = e1;

#pragma unroll
    for (int d = 0; d < 10; ++d) {
        float p = e0 * tv0[d] + e1 * tv1[d];
        for (int off = 16; off; off >>= 1) p += __shfl_xor(p, off, 32);
        if (lane == 0) attntmp[(long)ji * 16 + d] = p;
    }
    if (lane == 0)
#pragma unroll
        for (int d = 10; d < 16; ++d) attntmp[(long)ji * 16 + d] = 0.0f;
}

// ---------------------------------------------------------------------------

static inline void launch_gemm(hipStream_t s,
                               const float* A, const float* B, float* C, const float* bias,
                               int M, int N, int K,
                               int rsA, long bsA, int rsB, int csB, long bsB,
                               int ldC, long bsC, int batch, int accum)
{
    dim3 grid(N / 32, M / 16, batch);
    if (rsB == 1)
        hipLaunchKernelGGL((gemm_wmma_f32<true>), grid, dim3(32), 0, s,
                           A, B, C, bias, K, rsA, bsA, rsB, csB, bsB, ldC, bsC, accum);
    else
        hipLaunchKernelGGL((gemm_wmma_f32<false>), grid, dim3(32), 0, s,
                           A, B, C, bias, K, rsA, bsA, rsB, csB, bsB, ldC, bsC, accum);
}

extern "C" void kernel_launch(void* const* d_in, const int* in_sizes, int n_in,
                              void* d_out, int out_size, void* d_ws, size_t ws_size,
                              hipStream_t stream)
{
    const float* bd    = (const float*)d_in[0];   // (2048,256)
    const float* posi  = (const float*)d_in[1];   // (2,32,32,9)
    const float* temp  = (const float*)d_in[2];   // (64,64,32,10)
    const float* Wa    = (const float*)d_in[3];   // (256,64)
    const float* ba    = (const float*)d_in[4];   // (64)
    const float* Wb    = (const float*)d_in[5];   // (320,64)
    /* bb (d_in[6]) unused: softmax shift-invariance removes it exactly */
    const float* Wg    = (const float*)d_in[7];   // (320,256)
    const float* bg    = (const float*)d_in[8];   // (256)
    const float* Wpos  = (const float*)d_in[9];   // (9,32)
    const float* bpos  = (const float*)d_in[10];  // (32)
    const float* Wtmp  = (const float*)d_in[11];  // (10,64)
    const float* btmp  = (const float*)d_in[12];  // (64)
    float* out = (float*)d_out;                   // (64,32,256)

    float* w = (float*)d_ws;
    float* theta    = w;                  // 2048*64
    float* x_phi    = theta    + 131072;  // 2048*64
    float* x_g      = x_phi    + 131072;  // 2048*256
    float* posW     = x_g      + 524288;  // 1024*64
    float* posb     = posW     + 65536;   // 1024*64
    float* posg     = posb     + 65536;   // 1024*256
    float* WtB      = posg     + 262144;  // 32*64   (rows>=10 zero)
    float* WtG      = WtB      + 2048;    // 16*256  (rows>=10 zero)
    float* btG      = WtG      + 4096;    // 256
    float* thetaWtB = btG      + 256;     // 2048*32
    float* TXs      = thetaWtB + 65536;   // 64*32*32
    float* Pterm    = TXs      + 65536;   // 32*64*32
    float* Gterm    = Pterm    + 65536;   // 32*64*64
    float* attn_s   = Gterm    + 131072;  // 64*32*32
    float* attn_t   = attn_s   + 65536;   // 64*32*64
    float* attntmp  = attn_t   + 131072;  // 2048*16

    const float* WbP = Wb + 256 * 64;
    const float* WgP = Wg + 256 * 256;

    hipLaunchKernelGGL(precompute_w, dim3(1), dim3(256), 0, stream,
                       Wtmp, btmp, Wb, Wg, WtB, WtG, btG);
    hipLaunchKernelGGL(pos_project, dim3(256), dim3(256), 0, stream,
                       posi, Wpos, bpos, posW);

    // theta = x@Wa + ba            (2048,64,K=256)
    launch_gemm(stream, bd, Wa, theta, ba, 2048, 64, 256, 256, 0, 64, 1, 0, 64, 0, 1, 0);
    // x_phi = x@Wb[:256]           (2048,64,K=256)
    launch_gemm(stream, bd, Wb, x_phi, nullptr, 2048, 64, 256, 256, 0, 64, 1, 0, 64, 0, 1, 0);
    // x_g = x@Wg[:256] + bg        (2048,256,K=256)
    launch_gemm(stream, bd, Wg, x_g, bg, 2048, 256, 256, 256, 0, 256, 1, 0, 256, 0, 1, 0);
    // posb = posW@WbP              (1024,64,K=64)
    launch_gemm(stream, posW, WbP, posb, nullptr, 1024, 64, 64, 64, 0, 64, 1, 0, 64, 0, 1, 0);
    // posg = posW@WgP              (1024,256,K=64)
    launch_gemm(stream, posW, WgP, posg, nullptr, 1024, 256, 64, 64, 0, 256, 1, 0, 256, 0, 1, 0);
    // thetaWtB = theta@WtB^T       (2048,32,K=64): B[a][d] = WtB[d*64+a]
    launch_gemm(stream, theta, WtB, thetaWtB, nullptr, 2048, 32, 64, 64, 0, 1, 64, 0, 32, 0, 1, 0);
    // TXs[i] = theta[i]@x_phi[i]^T        (batch i=64; M=32,N=32,K=64)
    launch_gemm(stream, theta, x_phi, TXs, nullptr, 32, 32, 64, 64, 2048, 1, 64, 2048, 32, 1024, 64, 0);
    // Pterm[j] = theta[:,j,:]@posb[j]^T   (batch j=32; M=64,N=32,K=64)
    launch_gemm(stream, theta, posb, Pterm, nullptr, 64, 32, 64, 2048, 64, 1, 64, 2048, 32, 2048, 32, 0);
    // Gterm[i] = theta[:,i,:]@x_phi[:,i,:]^T  (batch spatial i=32; M=64,N=64,K=64)
    launch_gemm(stream, theta, x_phi, Gterm, nullptr, 64, 64, 64, 2048, 64, 1, 2048, 64, 64, 4096, 32, 0);

    hipLaunchKernelGGL(softmax_s, dim3(2048), dim3(32), 0, stream, TXs, Pterm, attn_s);
    hipLaunchKernelGGL(softmax_t, dim3(2048), dim3(32), 0, stream,
                       Gterm, temp, thetaWtB, attn_t, attntmp);

    // out[i] = attn_s[i]@x_g[i]            (batch i=64; M=32,N=256,K=32; write)
    launch_gemm(stream, attn_s, x_g, out, nullptr, 32, 256, 32, 32, 1024, 256, 1, 8192, 256, 8192, 64, 0);
    // out += attn_s[:,j,:]@posg[j]         (batch j=32; M=64,N=256,K=32; accumulate)
    launch_gemm(stream, attn_s, posg, out, nullptr, 64, 256, 32, 1024, 32, 256, 1, 8192, 8192, 256, 32, 1);
    // out += attn_t[:,i,:]@x_g[:,i,:]      (batch spatial i=32; M=64,N=256,K=64; accumulate)
    launch_gemm(stream, attn_t, x_g, out, nullptr, 64, 256, 64, 2048, 64, 8192, 1, 256, 8192, 256, 32, 1);
    // out += attntmp@WtG + btG             (M=2048,N=256,K=16; accumulate, bias)
    launch_gemm(stream, attntmp, WtG, out, btG, 2048, 256, 16, 16, 0, 256, 1, 0, 256, 0, 1, 1);
}